// Image_In_Attention_1889785610852
// MI455X (gfx1250) — compile-verified
//
#include <hip/hip_runtime.h>
#include <math.h>

#define B_ 256
#define S_ 196
#define R_ 1024
#define H_ 512

typedef float v2f __attribute__((ext_vector_type(2)));
typedef float v8f __attribute__((ext_vector_type(8)));

// ---------------------------------------------------------------------------
// Kernel 1: att_h[b][h] = dot(hidden[b][:], h2att_w[h][:]) + h2att_b[h]
// GEMM 256x512x1024 (f32) via V_WMMA_F32_16X16X4_F32.
// One wave32 per 16x16 output tile; EXEC all ones (blockDim == 32).
// ---------------------------------------------------------------------------
__global__ __launch_bounds__(32) void k_h2att_wmma(
    const float* __restrict__ hidden,   // [B, R]
    const float* __restrict__ w,        // [H, R]
    const float* __restrict__ bias,     // [H]
    float* __restrict__ att_h) {        // [B, H]
    const int tileB = blockIdx.x & 15;          // B/16 = 16 tiles
    const int tileH = blockIdx.x >> 4;          // H/16 = 32 tiles
    const int b0 = tileB * 16, h0 = tileH * 16;
    const int lane = threadIdx.x;               // 0..31
    const int mn   = lane & 15;                 // A: row m; B: col n
    const int half = lane >> 4;                 // K sub-pair selector

    const float* arow = hidden + (size_t)(b0 + mn) * R_;   // hidden row (M)
    const float* brow = w      + (size_t)(h0 + mn) * R_;   // weight row (N)

    v8f c = {};
    #pragma unroll 4
    for (int k0 = 0; k0 < R_; k0 += 4) {
        // A 16x4 layout: lanes 0-15 hold K={k0,k0+1}; lanes 16-31 K={k0+2,k0+3}
        v2f a = *(const v2f*)(arow + k0 + 2 * half);
        // B 4x16 layout: VGPR v holds row (v + 2*half), col = lane&15
        v2f b = *(const v2f*)(brow + k0 + 2 * half);
        c = __builtin_amdgcn_wmma_f32_16x16x4_f32(
                /*neg_a=*/false, a, /*neg_b=*/false, b,
                /*c_mod=*/(short)0, c, /*reuse_a=*/false, /*reuse_b=*/false);
    }

    const float bb = bias[h0 + mn];
    #pragma unroll
    for (int v = 0; v < 8; ++v) {
        const int m = v + 8 * half;             // C/D layout: row = vgpr + 8*half
        att_h[(size_t)(b0 + m) * H_ + (h0 + mn)] = c[v] + bb;
    }
}

// ---------------------------------------------------------------------------
// Kernel 2: scores[b][s] = sum_h tanh(in_w[b][s]*p[b][s][h] + att_h[b][h]) *
//                          alpha_w[h]   (+ alpha_b added here)
// One wave32 per (b,s); 4x float4 coalesced loads per lane cover H=512.
// ---------------------------------------------------------------------------
__global__ __launch_bounds__(256) void k_scores(
    const float* __restrict__ p_att,    // [B, S, H]
    const float* __restrict__ in_w,     // [B, S]
    const float* __restrict__ att_h,    // [B, H]
    const float* __restrict__ alpha_w,  // [H]
    const float* __restrict__ alpha_b,  // [1]
    float* __restrict__ scores) {       // [B, S]
    const int wave = threadIdx.x >> 5;
    const int lane = threadIdx.x & 31;
    const int pair = blockIdx.x * 8 + wave;     // 0 .. B*S-1 (exact multiple)
    const int b = pair / S_;

    const float  iw   = in_w[pair];
    const float* prow = p_att + (size_t)pair * H_;
    const float* hrow = att_h + (size_t)b * H_;

    float acc = 0.0f;
    #pragma unroll
    for (int i = 0; i < 4; ++i) {
        const int h = i * 128 + lane * 4;       // lanes cover 128 consecutive h
        float4 p  = *(const float4*)(prow + h);
        float4 ah = *(const float4*)(hrow + h);
        float4 aw = *(const float4*)(alpha_w + h);
        acc += tanhf(fmaf(iw, p.x, ah.x)) * aw.x;
        acc += tanhf(fmaf(iw, p.y, ah.y)) * aw.y;
        acc += tanhf(fmaf(iw, p.z, ah.z)) * aw.z;
        acc += tanhf(fmaf(iw, p.w, ah.w)) * aw.w;
    }
    #pragma unroll
    for (int off = 16; off > 0; off >>= 1)
        acc += __shfl_xor(acc, off, 32);
    if (lane == 0) scores[pair] = acc + alpha_b[0];
}

// ---------------------------------------------------------------------------
// Kernel 3: weight[b][:] = softmax(scores[b][:]) over S=196. One block per b.
// ---------------------------------------------------------------------------
__global__ __launch_bounds__(256) void k_softmax(
    const float* __restrict__ scores,   // [B, S]
    float* __restrict__ weight) {       // [B, S]
    __shared__ float red[256];
    const int b = blockIdx.x, t = threadIdx.x;

    float v = (t < S_) ? scores[b * S_ + t] : -INFINITY;
    red[t] = v;
    __syncthreads();
    #pragma unroll
    for (int o = 128; o > 0; o >>= 1) {
        if (t < o) red[t] = fmaxf(red[t], red[t + o]);
        __syncthreads();
    }
    const float m = red[0];
    __syncthreads();

    const float e = (t < S_) ? expf(v - m) : 0.0f;
    red[t] = e;
    __syncthreads();
    #pragma unroll
    for (int o = 128; o > 0; o >>= 1) {
        if (t < o) red[t] += red[t + o];
        __syncthreads();
    }
    const float inv = 1.0f / red[0];
    if (t < S_) weight[b * S_ + t] = e * inv;
}

// ---------------------------------------------------------------------------
// Kernel 4: att_res[b][r] = sum_s weight[b][s] * att_feats[b][s][r]
// One block per b; 256 threads x float4 = all 1024 r columns; coalesced
// 4KB row loads stream att_feats (205 MB) exactly once.
// ---------------------------------------------------------------------------
__global__ __launch_bounds__(256) void k_pool(
    const float* __restrict__ att_feats,  // [B, S, R]
    const float* __restrict__ weight,     // [B, S]
    float* __restrict__ att_res) {        // [B, R]
    __shared__ float w[S_];
    const int b = blockIdx.x, t = threadIdx.x;
    if (t < S_) w[t] = weight[b * S_ + t];
    __syncthreads();

    const float* base = att_feats + (size_t)b * S_ * R_ + t * 4;
    float4 acc = {0.f, 0.f, 0.f, 0.f};
    for (int s = 0; s < S_; ++s) {
        float4 v = *(const float4*)(base + (size_t)s * R_);
        const float ws = w[s];
        acc.x = fmaf(ws, v.x, acc.x);
        acc.y = fmaf(ws, v.y, acc.y);
        acc.z = fmaf(ws, v.z, acc.z);
        acc.w = fmaf(ws, v.w, acc.w);
    }
    *(float4*)(att_res + (size_t)b * R_ + t * 4) = acc;
}

// ---------------------------------------------------------------------------
extern "C" void kernel_launch(void* const* d_in, const int* in_sizes, int n_in,
                              void* d_out, int out_size, void* d_ws, size_t ws_size,
                              hipStream_t stream) {
    const float* att_feats = (const float*)d_in[0];  // [B,S,R]
    const float* p_att     = (const float*)d_in[1];  // [B,S,H]
    const float* hidden    = (const float*)d_in[2];  // [B,R]
    const float* in_w      = (const float*)d_in[3];  // [B,S]
    const float* h2att_w   = (const float*)d_in[4];  // [H,R]
    const float* h2att_b   = (const float*)d_in[5];  // [H]
    const float* alpha_w   = (const float*)d_in[6];  // [1,H]
    const float* alpha_b   = (const float*)d_in[7];  // [1]

    float* att_res = (float*)d_out;                        // [B,R]
    float* weight  = (float*)d_out + (size_t)B_ * R_;      // [B,S]

    float* att_h  = (float*)d_ws;                          // [B,H]  512 KB
    float* scores = att_h + (size_t)B_ * H_;               // [B,S]  200 KB

    k_h2att_wmma<<<(B_ / 16) * (H_ / 16), 32, 0, stream>>>(hidden, h2att_w, h2att_b, att_h);
    k_scores<<<(B_ * S_) / 8, 256, 0, stream>>>(p_att, in_w, att_h, alpha_w, alpha_b, scores);
    k_softmax<<<B_, 256, 0, stream>>>(scores, weight);
    k_pool<<<B_, 256, 0, stream>>>(att_feats, weight, att_res);
}